// SpikingLayer_13417477833295
// MI455X (gfx1250) — compile-verified
//
#include <hip/hip_runtime.h>
#include <stdint.h>

// ---------------- problem constants (from reference) ----------------
#define LIF_T     1024
#define LIF_B     32
#define LIF_N     1024
#define NSYN      2

#if defined(__HIP_DEVICE_COMPILE__)
#if __has_builtin(__builtin_amdgcn_tensor_load_to_lds) && __has_builtin(__builtin_amdgcn_s_wait_tensorcnt)
#define LIF_USE_TDM 1
#endif
#else
#define LIF_USE_TDM 1   // host pass: pick the TDM launch config
#endif

// =====================================================================
//  TDM path: one tensor_load_to_lds descriptor per chunk of 16 steps.
//  Block = 256 threads (8 waves) = 256 neurons of one batch; 128 blocks.
//  LDS: 2 buffers x (16 steps x 2 syn x 256 n x 4B) = 64 KB ring.
// =====================================================================
#if defined(LIF_USE_TDM)

#define TBLOCK   256
#define TCHUNK   16                          // time steps per TDM descriptor
#define TNCHUNK  (LIF_T / TCHUNK)            // 64
#define BUF_FLT  (TCHUNK * NSYN * TBLOCK)    // 8192 floats = 32 KB per buffer
#define CHUNK_GB (TCHUNK * NSYN * LIF_N * 4) // 131072 bytes of gmem per chunk

typedef unsigned int u32x4 __attribute__((ext_vector_type(4)));
typedef int          i32x8 __attribute__((ext_vector_type(8)));
typedef int          i32x4 __attribute__((ext_vector_type(4)));

#if defined(__HIP_DEVICE_COMPILE__)
__device__ __forceinline__ void tdm_issue_chunk(uint64_t gbase, uint32_t lds0, int c) {
  const uint64_t ga = gbase + (uint64_t)c * CHUNK_GB;        // tile start (bytes)
  const uint32_t la = lds0 + (uint32_t)(c & 1) * (BUF_FLT * 4);

  // ---- D# group 0 (128b): count / lds_addr / 57b global_addr / type=2 ----
  u32x4 g0;
  g0[0] = 1u;                                  // count=1 (valid), no gather
  g0[1] = la;                                  // lds_addr (bytes)
  g0[2] = (uint32_t)ga;                        // global_addr[31:0]
  g0[3] = (uint32_t)((ga >> 32) & 0x01FFFFFFu) // global_addr[56:32]
        | (2u << 30);                          // type = 2 ("image")

  // ---- D# group 1 (256b): sizes/strides.  Tile = 256(x) x 2(y) x 16(z),
  //      x contiguous, y stride 1024 elems (syn rows), z stride 2048 (t rows)
  i32x8 g1;
  g1[0] = (int)(2u << 16);                     // data_size=2 -> 4 bytes; mask=0
  g1[1] = (int)(256u << 16);                   // tensor_dim0[15:0]=256
  g1[2] = (int)(2u << 16);                     // tensor_dim1[15:0]=2
  g1[3] = (int)(256u << 16);                   // tile_dim0=256
  g1[4] = (int)((TCHUNK << 16) | 2u);          // tile_dim2=16, tile_dim1=2
  g1[5] = 1024;                                // tensor_dim0_stride[31:0]
  g1[6] = (int)(2048u << 16);                  // tensor_dim1_stride[15:0]
  g1[7] = 0;                                   // tensor_dim1_stride[47:16]

  // ---- D# group 2/3: tensor_dim2=16; dims 3/4 unused ----
  i32x4 g2;
  g2[0] = TCHUNK; g2[1] = 0; g2[2] = 0; g2[3] = 0;
  i32x4 g3;
  g3[0] = 0; g3[1] = 0; g3[2] = 0; g3[3] = 0;

  // This toolchain's builtin takes 6 args: the four D# SGPR groups
  // (VADDR0..VADDR3), an extra v8i32 (zero-filled; probe-verified), and cpol.
  i32x8 g4;
  g4[0] = 0; g4[1] = 0; g4[2] = 0; g4[3] = 0;
  g4[4] = 0; g4[5] = 0; g4[6] = 0; g4[7] = 0;

  __builtin_amdgcn_tensor_load_to_lds(g0, g1, g2, g3, g4, 0);
}
#endif // device

__global__ __launch_bounds__(TBLOCK)
void lif_scan_tdm_kernel(const float* __restrict__ in, float* __restrict__ out) {
#if defined(__HIP_DEVICE_COMPILE__)
  __shared__ float lbuf[2 * BUF_FLT];          // 64 KB double buffer

  const int tid = threadIdx.x;
  const int b   = blockIdx.x >> 2;             // 4 blocks per batch row
  const int n0  = (blockIdx.x & 3) << 8;       // neuron slice base
  const int n   = n0 + tid;

  const uint64_t gbase =
      (uint64_t)(uintptr_t)(in + ((size_t)b * LIF_T) * (NSYN * LIF_N) + n0);
  float* outp = out + (size_t)b * LIF_T * LIF_N + n;
  const uint32_t lds0 = (uint32_t)(uintptr_t)lbuf;   // low 32b of flat = DS addr

  // One wave issues DMA.  readfirstlane -> SGPR compare -> scalar branch:
  // tensor ops IGNORE EXEC, so we must branch, not predicate.
  const bool issuer = (__builtin_amdgcn_readfirstlane((int)threadIdx.x) == 0);

  if (issuer) tdm_issue_chunk(gbase, lds0, 0);

  float vmem = 0.0f, isyn0 = 0.0f, isyn1 = 0.0f;

  for (int c = 0; c < TNCHUNK; ++c) {
    __syncthreads();   // all waves finished reading buf[(c+1)&1] (chunk c-1)
    if (issuer) {
      if (c + 1 < TNCHUNK) {
        tdm_issue_chunk(gbase, lds0, c + 1);   // prefetch next chunk
        __builtin_amdgcn_s_wait_tensorcnt(1);  // in-order => chunk c landed
      } else {
        __builtin_amdgcn_s_wait_tensorcnt(0);
      }
    }
    __syncthreads();   // chunk c visible to all 8 waves

    // LDS layout from TDM fill order: [step][syn][n], packed.
    const float* lp = lbuf + (size_t)(c & 1) * BUF_FLT + tid;
    float* op = outp + (size_t)c * TCHUNK * LIF_N;
#pragma unroll
    for (int k = 0; k < TCHUNK; ++k) {
      const float in0 = lp[0];
      const float in1 = lp[256];
      // bitshift decay: x - x*2^-d == one rounding of x*(1-2^-d);
      // _rn intrinsics forbid FMA contraction (reference rounds mul/add apart)
      isyn0 = __fadd_rn(__fmul_rn(isyn0, 0.75f), in0);   // tau_syn=5 -> dash 2
      isyn1 = __fadd_rn(__fmul_rn(isyn1, 0.50f), in1);   // tau_syn=2 -> dash 1
      const float sum = __fadd_rn(isyn0, isyn1);
      vmem = __fadd_rn(__fmul_rn(vmem, 0.875f), sum);    // tau_mem=10 -> dash 3
      const float spk = (vmem >= 1.0f) ? 1.0f : 0.0f;    // spike BEFORE reset
      vmem = __fsub_rn(vmem, spk);                       // membrane subtract
      __builtin_nontemporal_store(spk, op);              // NT: spare the L2
      lp += NSYN * TBLOCK;
      op += LIF_N;
    }
  }
#endif // device
}

#else  // ------------- fallback: proven per-lane async-to-LDS path -----------

#define BLOCK     128
#define CHUNK     8
#define NBUF      4
#define NCHUNK    (LIF_T / CHUNK)
#define STEP_LDS  (BLOCK * NSYN * 4)
#define STEP_GMEM (NSYN * LIF_N * 4)
#define CHUNK_OPS (CHUNK * NSYN)

#define ASYNC_LOAD_B32(ldsaddr, gaddr)                                        \
  asm volatile("global_load_async_to_lds_b32 %0, %1, off"                     \
               :: "v"(ldsaddr), "v"(gaddr) : "memory")
#define WAIT_ASYNC(n)                                                         \
  asm volatile("s_wait_asynccnt %0" :: "n"(n) : "memory")

__global__ __launch_bounds__(BLOCK)
void lif_scan_async_kernel(const float* __restrict__ in, float* __restrict__ out) {
  __shared__ float lbuf[NBUF * CHUNK * BLOCK * NSYN];
  const int tid = threadIdx.x;
  const int gid = blockIdx.x * BLOCK + tid;
  const int b = gid >> 10;
  const int n = gid & (LIF_N - 1);
  const uint64_t in_base =
      (uint64_t)(uintptr_t)(in + ((size_t)b * LIF_T) * (NSYN * LIF_N) + n);
  float* outp = out + (size_t)b * LIF_T * LIF_N + n;
  const uint32_t lds_base = (uint32_t)(uintptr_t)lbuf + (uint32_t)tid * (NSYN * 4);
  float vmem = 0.0f, isyn0 = 0.0f, isyn1 = 0.0f;
  auto issue_chunk = [&](int c) {
    const int buf = c & (NBUF - 1);
    uint64_t g = in_base + (uint64_t)c * (CHUNK * STEP_GMEM);
    uint32_t l = lds_base + (uint32_t)(buf * CHUNK) * STEP_LDS;
#pragma unroll
    for (int k = 0; k < CHUNK; ++k) {
      ASYNC_LOAD_B32(l, g);
      ASYNC_LOAD_B32(l + 4u, g + 4096u);
      g += STEP_GMEM; l += STEP_LDS;
    }
  };
  issue_chunk(0); issue_chunk(1);
  for (int c = 0; c < NCHUNK; ++c) {
    if (c + 2 < NCHUNK)      { issue_chunk(c + 2); WAIT_ASYNC(2 * CHUNK_OPS); }
    else if (c + 1 < NCHUNK) { WAIT_ASYNC(CHUNK_OPS); }
    else                     { WAIT_ASYNC(0); }
    const int buf = c & (NBUF - 1);
    const float* lp = lbuf + (size_t)(buf * CHUNK) * (BLOCK * NSYN) + (size_t)tid * NSYN;
    float* op = outp + (size_t)c * CHUNK * LIF_N;
#pragma unroll
    for (int k = 0; k < CHUNK; ++k) {
      const float in0 = lp[0];
      const float in1 = lp[1];
      isyn0 = __fadd_rn(__fmul_rn(isyn0, 0.75f), in0);
      isyn1 = __fadd_rn(__fmul_rn(isyn1, 0.50f), in1);
      const float sum = __fadd_rn(isyn0, isyn1);
      vmem = __fadd_rn(__fmul_rn(vmem, 0.875f), sum);
      const float spk = (vmem >= 1.0f) ? 1.0f : 0.0f;
      vmem = __fsub_rn(vmem, spk);
      __builtin_nontemporal_store(spk, op);
      lp += BLOCK * NSYN; op += LIF_N;
    }
  }
}
#endif // LIF_USE_TDM

extern "C" void kernel_launch(void* const* d_in, const int* in_sizes, int n_in,
                              void* d_out, int out_size, void* d_ws, size_t ws_size,
                              hipStream_t stream) {
  (void)in_sizes; (void)n_in; (void)d_ws; (void)ws_size; (void)out_size;
  const float* in = (const float*)d_in[0];  // (B*T, n_syn, N) float32
  float* out = (float*)d_out;               // (B*T, N) float32
#if defined(LIF_USE_TDM)
  dim3 grid(LIF_B * LIF_N / TBLOCK);        // 128 blocks
  lif_scan_tdm_kernel<<<grid, dim3(TBLOCK), 0, stream>>>(in, out);
#else
  dim3 grid(LIF_B * LIF_N / BLOCK);         // 256 blocks
  lif_scan_async_kernel<<<grid, dim3(BLOCK), 0, stream>>>(in, out);
#endif
}